// Low_Rank_TBRNN_39290360824509
// MI455X (gfx1250) — compile-verified
//
#include <hip/hip_runtime.h>
#include <hip/hip_bf16.h>

typedef __attribute__((ext_vector_type(2))) float v2f;
typedef __attribute__((ext_vector_type(8))) float v8f;

#define B_ 64
#define T_ 1024
#define I_ 64
#define H_ 512
#define R_ 8
#define O_ 64
#define TAU_ 0.2f
#define STD_ 0.05f
#define SCALE_ (1.0f / (512.0f * 512.0f))

// ---------------------------------------------------------------------------
// Kernel A: inp[t,b,h] = sum_i u[b,t,i] * W_in[h,i] + b_in[h]
// GEMM rows = (t*B+b) over T*B, cols = h over H, K = I = 64.
// One wave per 16x16 output tile, fp32 WMMA 16x16x4, 16 K-steps.
// Memory-bound on the 134MB inp store; u/W_in reads are L2-resident.
// ---------------------------------------------------------------------------
__global__ __launch_bounds__(256) void inp_gemm_kernel(
    const float* __restrict__ u, const float* __restrict__ W_in,
    const float* __restrict__ b_in, float* __restrict__ inp)
{
    const int wave = threadIdx.x >> 5;
    const int lane = threadIdx.x & 31;
    const int tile = blockIdx.x * 8 + wave;       // 131072 tiles
    const int rowTile = tile >> 5;                // 4096 row tiles over T*B
    const int colTile = tile & 31;                // 32 col tiles over H
    const int r0 = rowTile << 4;
    const int t  = r0 >> 6;                       // row = t*B + b, 16 | B
    const int b0 = r0 & 63;
    const int h0 = colTile << 4;
    const int lm = lane & 15;                     // row/col within tile
    const int kk = lane >> 4;                     // selects K pair

    const float* ub = u    + ((size_t)(b0 + lm) * T_ + t) * I_ + 2 * kk;
    const float* wb = W_in + (size_t)(h0 + lm) * I_ + 2 * kk;

    v8f c = {};
#pragma unroll
    for (int k = 0; k < I_ / 4; ++k) {
        v2f a  = *(const v2f*)(ub + 4 * k);
        v2f bm = *(const v2f*)(wb + 4 * k);
        c = __builtin_amdgcn_wmma_f32_16x16x4_f32(false, a, false, bm,
                                                  (short)0, c, false, false);
    }
    const float bin = b_in[h0 + lm];
    float* ob = inp + (size_t)t * (B_ * H_) + (size_t)(b0 + 8 * kk) * H_ + h0 + lm;
#pragma unroll
    for (int v = 0; v < 8; ++v)
        ob[(size_t)v * H_] = c[v] + bin;
}

// ---------------------------------------------------------------------------
// Kernel B: sequential scan. One workgroup per batch (independent chain,
// no global sync ever). x held in registers (2 h-elements per thread),
// M/N/L rows in registers. Rank-16 partial reduction via a value-halving
// wave32 butterfly (8+4+2+1+1 exchanges instead of 5*16), then a tiny
// cross-wave LDS combine. noise/inp prefetched one step ahead.
// ---------------------------------------------------------------------------
__global__ __launch_bounds__(256) void scan_kernel(
    const float* __restrict__ inp, const float* __restrict__ noise,
    const float* __restrict__ x0, const float* __restrict__ L,
    const float* __restrict__ M, const float* __restrict__ N,
    float* __restrict__ traj, float* __restrict__ x_last)
{
    __shared__ float sv[8][16];   // [wave][j]: j 0..7 = r@M, 8..15 = r@N
    __shared__ float qs[8];

    const int b    = blockIdx.x;
    const int tid  = threadIdx.x;
    const int wave = tid >> 5;
    const int lane = tid & 31;
    const int h0   = tid * 2;

    float Mr[2][8], Nr[2][8], Lr[2][8];
#pragma unroll
    for (int j = 0; j < 8; ++j) {
        Mr[0][j] = M[(size_t)h0 * 8 + j];  Mr[1][j] = M[(size_t)(h0 + 1) * 8 + j];
        Nr[0][j] = N[(size_t)h0 * 8 + j];  Nr[1][j] = N[(size_t)(h0 + 1) * 8 + j];
        Lr[0][j] = L[(size_t)h0 * 8 + j];  Lr[1][j] = L[(size_t)(h0 + 1) * 8 + j];
    }

    // destination LDS slot for this lane's butterfly result: j = bitrev4(lane&15)
    const int jdst = ((lane & 1) << 3) | ((lane & 2) << 1) |
                     ((lane & 4) >> 1) | ((lane & 8) >> 3);

    float x0v = x0[(size_t)b * H_ + h0];
    float x1v = x0[(size_t)b * H_ + h0 + 1];

    const size_t BH = (size_t)B_ * H_;
    const float* np = noise + (size_t)b * H_ + h0;   // + t*B*H per step
    const float* ip = inp   + (size_t)b * H_ + h0;
    float*       tp = traj  + (size_t)b * T_ * H_ + h0;

    // prime the prefetch pipeline (t = 0)
    float nz0 = __builtin_nontemporal_load(np);
    float nz1 = __builtin_nontemporal_load(np + 1);
    float iv0 = ip[0];
    float iv1 = ip[1];

    for (int t = 0; t < T_; ++t) {
        float nn0 = 0.f, nn1 = 0.f, in0 = 0.f, in1 = 0.f;
        if (t + 1 < T_) {                       // prefetch next step's operands
            const float* np2 = np + (size_t)(t + 1) * BH;
            const float* ip2 = ip + (size_t)(t + 1) * BH;
            nn0 = __builtin_nontemporal_load(np2);
            nn1 = __builtin_nontemporal_load(np2 + 1);
            in0 = ip2[0];
            in1 = ip2[1];
        }

        const float r0 = tanhf(x0v);
        const float r1 = tanhf(x1v);

        float vals[16];
#pragma unroll
        for (int j = 0; j < 8; ++j) {
            vals[j]     = r0 * Mr[0][j] + r1 * Mr[1][j];
            vals[j + 8] = r0 * Nr[0][j] + r1 * Nr[1][j];
        }

        // value-halving butterfly: after 4 levels each lane holds one fully
        // (16-lane) reduced value; level 5 (xor 16) completes the 32-lane sum.
        {
            const bool hi = (lane & 1) != 0;
#pragma unroll
            for (int k = 0; k < 8; ++k) {
                float send = hi ? vals[k] : vals[k + 8];
                float keep = hi ? vals[k + 8] : vals[k];
                vals[k] = keep + __shfl_xor(send, 1, 32);
            }
        }
        {
            const bool hi = (lane & 2) != 0;
#pragma unroll
            for (int k = 0; k < 4; ++k) {
                float send = hi ? vals[k] : vals[k + 4];
                float keep = hi ? vals[k + 4] : vals[k];
                vals[k] = keep + __shfl_xor(send, 2, 32);
            }
        }
        {
            const bool hi = (lane & 4) != 0;
#pragma unroll
            for (int k = 0; k < 2; ++k) {
                float send = hi ? vals[k] : vals[k + 2];
                float keep = hi ? vals[k + 2] : vals[k];
                vals[k] = keep + __shfl_xor(send, 4, 32);
            }
        }
        {
            const bool hi = (lane & 8) != 0;
            float send = hi ? vals[0] : vals[1];
            float keep = hi ? vals[1] : vals[0];
            vals[0] = keep + __shfl_xor(send, 8, 32);
        }
        vals[0] += __shfl_xor(vals[0], 16, 32);

        if (lane < 16)
            sv[wave][jdst] = vals[0];
        __syncthreads();

        if (tid < 8) {
            float am = 0.f, an = 0.f;
#pragma unroll
            for (int w = 0; w < 8; ++w) { am += sv[w][tid]; an += sv[w][tid + 8]; }
            qs[tid] = am * an * SCALE_;         // ((r@M)*(r@N))[b,j] * scale
        }
        __syncthreads();

        float rec0 = 0.f, rec1 = 0.f;
#pragma unroll
        for (int j = 0; j < 8; ++j) {
            const float q = qs[j];
            rec0 += q * Lr[0][j];
            rec1 += q * Lr[1][j];
        }

        x0v += STD_ * nz0 + TAU_ * (-x0v + rec0 + iv0);
        x1v += STD_ * nz1 + TAU_ * (-x1v + rec1 + iv1);

        float* tpt = tp + (size_t)t * H_;
        tpt[0] = x0v;
        tpt[1] = x1v;

        nz0 = nn0; nz1 = nn1; iv0 = in0; iv1 = in1;
    }

    x_last[(size_t)b * H_ + h0]     = x0v;
    x_last[(size_t)b * H_ + h0 + 1] = x1v;
}

// ---------------------------------------------------------------------------
// Kernel C: out[b,t,o] = sum_h tanh(traj[b,t,h]) * W_out[o,h] + b_out[o]
// Each WG owns 32 rows of B*T; stages tanh(traj[32][512]) into LDS ONCE
// (padded pitch -> conflict-free fragment reads), then its 8 waves compute
// the 2 row-tiles x 4 col-tiles of 16x16 output with fp32 WMMA, A-fragments
// from LDS, B-fragments (W_out, 128KB, L2-hot) from global.
// ---------------------------------------------------------------------------
#define APITCH 516   // 512 + 4-float pad: row->row bank offset of 4
__global__ __launch_bounds__(256) void out_gemm_kernel(
    const float* __restrict__ traj, const float* __restrict__ W_out,
    const float* __restrict__ b_out, float* __restrict__ out)
{
    __shared__ float As[32 * APITCH];            // 66 KB of the 320 KB LDS

    const int tid  = threadIdx.x;
    const int wave = tid >> 5;
    const int lane = tid & 31;
    const int r0   = blockIdx.x << 5;            // 2048 blocks * 32 rows = B*T

    // stage tanh(traj) for 32 rows: one row per iteration, float2 per thread
#pragma unroll 4
    for (int row = 0; row < 32; ++row) {
        v2f a = *(const v2f*)(traj + (size_t)(r0 + row) * H_ + 2 * tid);
        a.x = tanhf(a.x);
        a.y = tanhf(a.y);
        *(v2f*)(&As[row * APITCH + 2 * tid]) = a;
    }
    __syncthreads();

    const int rowSub = wave >> 2;                // 0..1
    const int o0     = (wave & 3) << 4;          // 0..48
    const int lm     = lane & 15;
    const int kk     = lane >> 4;

    const float* ab = &As[(rowSub * 16 + lm) * APITCH + 2 * kk];
    const float* wb = W_out + (size_t)(o0 + lm) * H_ + 2 * kk;

    v8f c = {};
#pragma unroll 8
    for (int k = 0; k < H_ / 4; ++k) {
        v2f a  = *(const v2f*)(ab + 4 * k);      // ds_load_b64, conflict-free
        v2f bm = *(const v2f*)(wb + 4 * k);
        c = __builtin_amdgcn_wmma_f32_16x16x4_f32(false, a, false, bm,
                                                  (short)0, c, false, false);
    }
    const float bo = b_out[o0 + lm];
    float* ob = out + (size_t)(r0 + rowSub * 16 + 8 * kk) * O_ + o0 + lm;
#pragma unroll
    for (int v = 0; v < 8; ++v)
        ob[(size_t)v * O_] = c[v] + bo;
}

// ---------------------------------------------------------------------------
extern "C" void kernel_launch(void* const* d_in, const int* in_sizes, int n_in,
                              void* d_out, int out_size, void* d_ws, size_t ws_size,
                              hipStream_t stream)
{
    const float* u     = (const float*)d_in[0];  // [B,T,I]
    const float* x0    = (const float*)d_in[1];  // [B,H]
    const float* noise = (const float*)d_in[2];  // [T,B,H]
    const float* L     = (const float*)d_in[3];  // [H,R]
    const float* M     = (const float*)d_in[4];  // [H,R]
    const float* N     = (const float*)d_in[5];  // [H,R]
    const float* W_in  = (const float*)d_in[6];  // [H,I]
    const float* b_in  = (const float*)d_in[7];  // [H]
    const float* W_out = (const float*)d_in[8];  // [O,H]
    const float* b_out = (const float*)d_in[9];  // [O]

    float* out    = (float*)d_out;                         // [B,T,O]
    float* x_last = out + (size_t)B_ * T_ * O_;            // [B,H]
    float* traj   = x_last + (size_t)B_ * H_;              // [B,T,H]
    float* inp    = (float*)d_ws;                          // [T,B,H] scratch (128 MiB)

    // Phase 1: input-drive GEMM (parallel, memory-bound; inp stays L2-hot)
    inp_gemm_kernel<<<16384, 256, 0, stream>>>(u, W_in, b_in, inp);
    // Phase 2: 64 independent sequential chains, one WG per batch
    scan_kernel<<<64, 256, 0, stream>>>(inp, noise, x0, L, M, N, traj, x_last);
    // Phase 3: readout GEMM over tanh(traj), LDS-staged A
    out_gemm_kernel<<<2048, 256, 0, stream>>>(traj, W_out, b_out, out);
}